// ParallelHeadFusion_25185688224040
// MI455X (gfx1250) — compile-verified
//
#include <hip/hip_runtime.h>
#include <math.h>

#define N_NODES 20000
#define E_EDGES 320000
#define ETOT (E_EDGES + N_NODES)
#define DIN 128
#define NHEAD 4
#define FCH 128
#define NEG_SLOPE 0.2f
#define MT 5   // M-tiles per wave: 20000/16 = 1250 = 5*250 (exact)

typedef __attribute__((ext_vector_type(2))) float v2f;
typedef __attribute__((ext_vector_type(8))) float v8f;

// ---------------------------------------------------------------------------
// WMMA f32 GEMM: C[M,N] = act(A[M,K] @ B[K,N] + bias)
// Each wave owns MT vertically-stacked 16x16 C tiles in one N column strip:
// one B fragment per k-step feeds MT independent V_WMMA_F32_16X16X4_F32
// accumulations. Loads are issued as a batch BEFORE the WMMA group so the
// compiler can clause them and stagger s_wait_loadcnt instead of draining
// to zero before every WMMA.
// A layout (ISA 7.12.2, 32-bit A 16x4): lane m=lane&15 holds row m;
//   lanes 0-15 carry K+0,K+1 in the 2 VGPRs, lanes 16-31 carry K+2,K+3.
// B mirrored: lane n=lane&15 holds col n; lane-half selects K pair.
// C/D: VGPR r, lanes0-15 -> row r, lanes16-31 -> row r+8.
// All guards are wave-uniform => EXEC all-1s at every WMMA.
// ---------------------------------------------------------------------------
__global__ void gemm_wmma_f32(const float* __restrict__ A, int lda,
                              const float* __restrict__ B, int ldb,
                              const float* __restrict__ bias,
                              float* __restrict__ C, int ldc,
                              int M, int N, int K, int relu)
{
    const int lane = threadIdx.x & 31;
    const int wave = threadIdx.x >> 5;
    const int tiles_n = N >> 4;
    const int tiles_m = M >> 4;
    const int macro_m = (tiles_m + MT - 1) / MT;
    const int w = blockIdx.x * 4 + wave;            // wave-uniform
    if (w >= macro_m * tiles_n) return;             // whole wave exits together
    const int tm0 = (w / tiles_n) * MT;             // first m-tile of this wave
    const int n0  = (w % tiles_n) << 4;
    const int lidx = lane & 15;
    const int hi   = lane >> 4;                     // 0/1: selects K pair

    // Per-subtile A row pointers (clamped uniformly; stores are guarded).
    const float* Arow[MT];
#pragma unroll
    for (int i = 0; i < MT; ++i) {
        int ti = tm0 + i; if (ti >= tiles_m) ti = tiles_m - 1;
        Arow[i] = A + (size_t)((ti << 4) + lidx) * lda + (hi << 1);
    }
    const float* Bcol = B + (size_t)(hi << 1) * ldb + (n0 + lidx);

    v8f acc[MT];
#pragma unroll
    for (int i = 0; i < MT; ++i) acc[i] = (v8f){};

    for (int k = 0; k < K; k += 4) {
        // ---- phase 1: issue all loads for this k-step (clause-able) ----
        float bx = Bcol[(size_t)k * ldb];
        float by = Bcol[(size_t)(k + 1) * ldb];
        float2 av[MT];
#pragma unroll
        for (int i = 0; i < MT; ++i)
            av[i] = *(const float2*)(Arow[i] + k);  // contiguous K pair

        // ---- phase 2: MT independent WMMAs sharing the B fragment ----
        v2f b; b.x = bx; b.y = by;
#pragma unroll
        for (int i = 0; i < MT; ++i) {
            v2f a; a.x = av[i].x; a.y = av[i].y;
            acc[i] = __builtin_amdgcn_wmma_f32_16x16x4_f32(
                false, a, false, b, (short)0, acc[i], false, false);
        }
    }

    const int col = n0 + lidx;
    const float bv = bias ? bias[col] : 0.0f;
#pragma unroll
    for (int i = 0; i < MT; ++i) {
        if (tm0 + i >= tiles_m) break;              // wave-uniform
        const int mbase = (tm0 + i) << 4;
#pragma unroll
        for (int r = 0; r < 8; ++r) {
            const int row = mbase + r + (hi << 3);
            float v = acc[i][r] + bv;
            if (relu) v = fmaxf(v, 0.0f);
            C[(size_t)row * ldc + col] = v;
        }
    }
}

// ---------------------------------------------------------------------------
// Ordered-uint encoding for float atomicMax (memset-0 init < any real code).
// ---------------------------------------------------------------------------
__device__ __forceinline__ unsigned f2ord(float f) {
    unsigned u = __float_as_uint(f);
    return (u & 0x80000000u) ? ~u : (u | 0x80000000u);
}
__device__ __forceinline__ float ord2f(unsigned u) {
    return (u & 0x80000000u) ? __uint_as_float(u & 0x7FFFFFFFu)
                             : __uint_as_float(~u);
}

__global__ void reduce_sum_kernel(const float* __restrict__ v, int n,
                                  float* __restrict__ out)
{
    float s = 0.0f;
    for (int i = blockIdx.x * blockDim.x + threadIdx.x; i < n;
         i += gridDim.x * blockDim.x)
        s += v[i];
    for (int o = 16; o > 0; o >>= 1) s += __shfl_down(s, o, 32);
    if ((threadIdx.x & 31) == 0) atomicAdd(out, s);
}

// logits[e,h] = sum_c att[h,c] * leaky_relu(xl[src,h,c] + xr[dst,h,c] + ea*We[h,c])
// All row bases are 16B aligned (oc multiple of 16) -> float4 loads.
__global__ void edge_logits_kernel(const float* __restrict__ xl,
                                   const float* __restrict__ xr,
                                   const float* __restrict__ ea,
                                   const int* __restrict__ srcA,
                                   const int* __restrict__ dstA,
                                   const float* __restrict__ We,
                                   const float* __restrict__ att,
                                   const float* __restrict__ ea_sum,
                                   float* __restrict__ logits, int oc)
{
    const int t = blockIdx.x * blockDim.x + threadIdx.x;
    if (t >= ETOT * NHEAD) return;
    const int e = t >> 2, h = t & 3;
    int s, d; float eav;
    if (e < E_EDGES) { s = srcA[e]; d = dstA[e]; eav = ea[e]; }
    else             { s = e - E_EDGES; d = s; eav = ea_sum[0] * (1.0f / E_EDGES); }
    const int HC = NHEAD * oc;
    const float4* xls  = (const float4*)(xl  + (size_t)s * HC + h * oc);
    const float4* xrd  = (const float4*)(xr  + (size_t)d * HC + h * oc);
    const float4* Weh  = (const float4*)(We  + h * oc);
    const float4* atth = (const float4*)(att + h * oc);
    float acc = 0.0f;
    const int q = oc >> 2;
    for (int c = 0; c < q; ++c) {
        const float4 l = xls[c], r = xrd[c], w = Weh[c], a = atth[c];
        float v;
        v = l.x + r.x + eav * w.x; v = (v > 0.0f) ? v : NEG_SLOPE * v; acc += v * a.x;
        v = l.y + r.y + eav * w.y; v = (v > 0.0f) ? v : NEG_SLOPE * v; acc += v * a.y;
        v = l.z + r.z + eav * w.z; v = (v > 0.0f) ? v : NEG_SLOPE * v; acc += v * a.z;
        v = l.w + r.w + eav * w.w; v = (v > 0.0f) ? v : NEG_SLOPE * v; acc += v * a.w;
    }
    logits[t] = acc;
}

__global__ void seg_max_kernel(const float* __restrict__ logits,
                               const int* __restrict__ dstA,
                               unsigned* __restrict__ mxu)
{
    const int t = blockIdx.x * blockDim.x + threadIdx.x;
    if (t >= ETOT * NHEAD) return;
    const int e = t >> 2, h = t & 3;
    const int d = (e < E_EDGES) ? dstA[e] : (e - E_EDGES);
    atomicMax(&mxu[d * NHEAD + h], f2ord(logits[t]));
}

__global__ void seg_exp_kernel(const float* __restrict__ logits,
                               const int* __restrict__ dstA,
                               const unsigned* __restrict__ mxu,
                               float* __restrict__ den,
                               float* __restrict__ alpha)
{
    const int t = blockIdx.x * blockDim.x + threadIdx.x;
    if (t >= ETOT * NHEAD) return;
    const int e = t >> 2, h = t & 3;
    const int d = (e < E_EDGES) ? dstA[e] : (e - E_EDGES);
    const float ex = expf(logits[t] - ord2f(mxu[d * NHEAD + h]));
    alpha[t] = ex;
    atomicAdd(&den[d * NHEAD + h], ex);
}

// alpha normalize + scatter: outacc[d,h,c] += alpha * xl[s,h,c]
__global__ void alpha_scatter_kernel(const int* __restrict__ srcA,
                                     const int* __restrict__ dstA,
                                     const float* __restrict__ den,
                                     const float* __restrict__ xl,
                                     float* __restrict__ alpha,
                                     float* __restrict__ outacc, int oc)
{
    const int t = blockIdx.x * blockDim.x + threadIdx.x;
    if (t >= ETOT * NHEAD) return;
    const int e = t >> 2, h = t & 3;
    int s, d;
    if (e < E_EDGES) { s = srcA[e]; d = dstA[e]; }
    else             { s = e - E_EDGES; d = s; }
    const float a = alpha[t] / den[d * NHEAD + h];
    alpha[t] = a;                                   // final alpha output
    const int HC = NHEAD * oc;
    const float4* xls = (const float4*)(xl + (size_t)s * HC + h * oc);
    float* oa = outacc + (size_t)d * HC + h * oc;
    const int q = oc >> 2;
    for (int c = 0; c < q; ++c) {
        const float4 l = xls[c];
        atomicAdd(&oa[4 * c + 0], a * l.x);
        atomicAdd(&oa[4 * c + 1], a * l.y);
        atomicAdd(&oa[4 * c + 2], a * l.z);
        atomicAdd(&oa[4 * c + 3], a * l.w);
    }
}

// hnode[n,c] = relu(mean_h(outacc[n,h,c]) + bias[c]) + skip[n,c]
__global__ void node_finalize_kernel(const float* __restrict__ outacc,
                                     const float* __restrict__ bias,
                                     const float* __restrict__ skip,
                                     float* __restrict__ hnode, int oc)
{
    const int t = blockIdx.x * blockDim.x + threadIdx.x;
    if (t >= N_NODES * oc) return;
    const int n = t / oc, c = t % oc;
    const float* oa = outacc + (size_t)n * NHEAD * oc + c;
    float g = 0.0f;
#pragma unroll
    for (int h = 0; h < NHEAD; ++h) g += oa[h * oc];
    g = g * (1.0f / NHEAD) + bias[c];
    hnode[t] = fmaxf(g, 0.0f) + skip[t];
}

__global__ void edge_fuse_kernel(const float* __restrict__ ea,
                                 const float* __restrict__ Wef,
                                 const float* __restrict__ bef,
                                 float* __restrict__ out)
{
    const int e = blockIdx.x * blockDim.x + threadIdx.x;
    if (e >= E_EDGES) return;
    const float v = ea[e] * (Wef[0] + Wef[1] + Wef[2]) + bef[0];
    out[e] = fmaxf(v, 0.0f);
}

static inline void launch_gemm(const float* A, int lda, const float* B, int ldb,
                               const float* bias, float* C, int ldc,
                               int M, int N, int K, int relu, hipStream_t stream)
{
    const int tiles_m = M >> 4;
    const int macro_m = (tiles_m + MT - 1) / MT;
    const int waves = macro_m * (N >> 4);
    const int blocks = (waves + 3) >> 2;             // 4 waves per 128-thread block
    gemm_wmma_f32<<<blocks, 128, 0, stream>>>(A, lda, B, ldb, bias, C, ldc,
                                              M, N, K, relu);
}

extern "C" void kernel_launch(void* const* d_in, const int* in_sizes, int n_in,
                              void* d_out, int out_size, void* d_ws, size_t ws_size,
                              hipStream_t stream)
{
    (void)in_sizes; (void)n_in; (void)out_size; (void)ws_size;
    const float* x    = (const float*)d_in[0];
    const int*   eidx = (const int*)d_in[1];
    const float* ea   = (const float*)d_in[2];
    // d_in[3] = batch (all zeros, unused)
    const int OCS[3] = {16, 32, 64};

    // ---- workspace carve-up (float units, 256B-aligned offsets) ----
    float* ws = (float*)d_ws;
    size_t off = 0;
    float*    ea_sum = ws + off; off += 64;
    float*    xl     = ws + off; off += (size_t)N_NODES * 256;
    float*    xr     = ws + off; off += (size_t)N_NODES * 256;
    float*    skip   = ws + off; off += (size_t)N_NODES * 64;
    float*    logits = ws + off; off += (size_t)ETOT * NHEAD;
    unsigned* mxu    = (unsigned*)(ws + off); off += (size_t)N_NODES * NHEAD;
    float*    den    = ws + off; off += (size_t)N_NODES * NHEAD;
    float*    outacc = ws + off; off += (size_t)N_NODES * 256;
    float*    hnode  = ws + off; off += (size_t)N_NODES * 64;
    float*    cat    = ws + off; off += (size_t)N_NODES * 384;

    float* x_fused   = (float*)d_out;
    float* ea_fused  = x_fused + (size_t)N_NODES * FCH;
    float* alpha0    = ea_fused + E_EDGES;

    const int* srcA = eidx;             // edge_index[0,:]
    const int* dstA = eidx + E_EDGES;   // edge_index[1,:]

    // edge_attr mean (fill value for self loops)
    hipMemsetAsync(ea_sum, 0, sizeof(float), stream);
    reduce_sum_kernel<<<256, 256, 0, stream>>>(ea, E_EDGES, ea_sum);

    const int et = ETOT * NHEAD;
    const int eblocks = (et + 255) / 256;

    for (int b = 0; b < 3; ++b) {
        const int oc = OCS[b];
        const int HC = NHEAD * oc;
        const float* Wl    = (const float*)d_in[4 + b * 10 + 0];
        const float* bl    = (const float*)d_in[4 + b * 10 + 1];
        const float* Wr    = (const float*)d_in[4 + b * 10 + 2];
        const float* br    = (const float*)d_in[4 + b * 10 + 3];
        const float* We    = (const float*)d_in[4 + b * 10 + 4];
        const float* att   = (const float*)d_in[4 + b * 10 + 5];
        const float* bias  = (const float*)d_in[4 + b * 10 + 6];
        const float* Wskip = (const float*)d_in[4 + b * 10 + 7];
        const float* Wlin  = (const float*)d_in[4 + b * 10 + 8];
        const float* blin  = (const float*)d_in[4 + b * 10 + 9];

        // dense transforms (WMMA)
        launch_gemm(x, DIN, Wl,    HC, bl,      xl,   HC, N_NODES, HC, DIN, 0, stream);
        launch_gemm(x, DIN, Wr,    HC, br,      xr,   HC, N_NODES, HC, DIN, 0, stream);
        launch_gemm(x, DIN, Wskip, oc, nullptr, skip, oc, N_NODES, oc, DIN, 0, stream);

        hipMemsetAsync(mxu,    0, (size_t)N_NODES * NHEAD * sizeof(unsigned), stream);
        hipMemsetAsync(den,    0, (size_t)N_NODES * NHEAD * sizeof(float), stream);
        hipMemsetAsync(outacc, 0, (size_t)N_NODES * HC * sizeof(float), stream);

        float* alpha = alpha0 + (size_t)b * ETOT * NHEAD;
        edge_logits_kernel<<<eblocks, 256, 0, stream>>>(xl, xr, ea, srcA, dstA,
                                                        We, att, ea_sum, logits, oc);
        seg_max_kernel<<<eblocks, 256, 0, stream>>>(logits, dstA, mxu);
        seg_exp_kernel<<<eblocks, 256, 0, stream>>>(logits, dstA, mxu, den, alpha);
        alpha_scatter_kernel<<<eblocks, 256, 0, stream>>>(srcA, dstA, den, xl,
                                                          alpha, outacc, oc);

        const int ntc = N_NODES * oc;
        node_finalize_kernel<<<(ntc + 255) / 256, 256, 0, stream>>>(outacc, bias,
                                                                    skip, hnode, oc);
        // per-block projection into the concat buffer (column offset b*FCH)
        launch_gemm(hnode, oc, Wlin, FCH, blin, cat + b * FCH, 384,
                    N_NODES, FCH, oc, 0, stream);
    }

    const float* Wf  = (const float*)d_in[34];
    const float* bf  = (const float*)d_in[35];
    const float* Wef = (const float*)d_in[36];
    const float* bef = (const float*)d_in[37];

    launch_gemm(cat, 384, Wf, FCH, bf, x_fused, FCH, N_NODES, FCH, 384, 1, stream);
    edge_fuse_kernel<<<(E_EDGES + 255) / 256, 256, 0, stream>>>(ea, Wef, bef, ea_fused);
}